// LSTM_4148938408391
// MI455X (gfx1250) — compile-verified
//
#include <hip/hip_runtime.h>

// LSTM: B=16384 seqs, T=1024 steps, I=1, H=2, C=4.
// One lane per batch element; 512 waves total (parallelism cap = B/32) ->
// runtime == per-wave issue count. Optimizations:
//  * native v_tanh_f32 (1 TRANS per nonlinearity)
//  * sigmoid pre-scale folded into i/f/o gate weights (kills 6 muls/step)
//  * packed V_PK_FMA_F32 gate accumulation (24 fma -> 12 pk_fma/step)
//  * double-buffered global_load_b128 x-chunks + global_prefetch_b8
//  * epilogue FC via V_WMMA_F32_16X16X4_F32 (K=2 zero-padded to 4)

#define T_LEN 1024
#define HID   2
#define NCLS  4
#define ROWS_PER_BLOCK 64   // 2 waves per block
#define CHUNK 32            // one 128B line of x per chunk

typedef __attribute__((ext_vector_type(2))) float v2f;
typedef __attribute__((ext_vector_type(8))) float v8f;

__device__ __forceinline__ float tanh_fast(float x) {
#if __has_builtin(__builtin_amdgcn_tanhf)
  return __builtin_amdgcn_tanhf(x);
#elif __has_builtin(__builtin_amdgcn_tanh_f32)
  return __builtin_amdgcn_tanh_f32(x);
#else
  float r;
  asm("v_tanh_f32 %0, %1" : "=v"(r) : "v"(x));
  return r;
#endif
}

// For pre-scaled accumulators: sigmoid(2y) = 0.5*tanh(y)+0.5
__device__ __forceinline__ float sig_post(float y) {
  return fmaf(0.5f, tanh_fast(y), 0.5f);
}

__device__ __forceinline__ v2f v2fma(v2f a, v2f b, v2f c) {
  return __builtin_elementwise_fma(a, b, c);
}

__device__ __forceinline__ void load_chunk(float (&buf)[CHUNK], const float* xr, int t0) {
  const float4* p = (const float4*)(xr + t0);
#pragma unroll
  for (int i = 0; i < CHUNK / 4; ++i) {
    float4 v = p[i];
    buf[4 * i + 0] = v.x;
    buf[4 * i + 1] = v.y;
    buf[4 * i + 2] = v.z;
    buf[4 * i + 3] = v.w;
  }
}

struct State { v2f h, c; };

// Gate order (PyTorch): 0=i 1=f 2=g 3=o ; each gate is a v2f over hidden units.
__device__ __forceinline__ void lstm_steps(const float (&buf)[CHUNK],
                                           const v2f (&wih)[4],
                                           const v2f (&bias)[4],
                                           const v2f (&whh0)[4],
                                           const v2f (&whh1)[4],
                                           State& s) {
#pragma unroll
  for (int k = 0; k < CHUNK; ++k) {
    float xt = buf[k];
    v2f xs  = {xt, xt};
    v2f h0s = __builtin_shufflevector(s.h, s.h, 0, 0);  // broadcast h0 (OPSEL-foldable)
    v2f h1s = __builtin_shufflevector(s.h, s.h, 1, 1);  // broadcast h1

    v2f ai = v2fma(h1s, whh1[0], v2fma(h0s, whh0[0], v2fma(xs, wih[0], bias[0])));
    v2f af = v2fma(h1s, whh1[1], v2fma(h0s, whh0[1], v2fma(xs, wih[1], bias[1])));
    v2f ag = v2fma(h1s, whh1[2], v2fma(h0s, whh0[2], v2fma(xs, wih[2], bias[2])));
    v2f ao = v2fma(h1s, whh1[3], v2fma(h0s, whh0[3], v2fma(xs, wih[3], bias[3])));

    float i0 = sig_post(ai.x), i1 = sig_post(ai.y);   // i/f/o rows pre-scaled by 0.5
    float f0 = sig_post(af.x), f1 = sig_post(af.y);
    float g0 = tanh_fast(ag.x), g1 = tanh_fast(ag.y);
    float o0 = sig_post(ao.x), o1 = sig_post(ao.y);

    v2f fv = {f0, f1};
    v2f ig = {i0 * g0, i1 * g1};
    s.c = v2fma(fv, s.c, ig);
    s.h = (v2f){o0 * tanh_fast(s.c.x), o1 * tanh_fast(s.c.y)};
  }
}

__global__ __launch_bounds__(ROWS_PER_BLOCK)
void lstm_fused_kernel(const float* __restrict__ x,
                       const float* __restrict__ W_ih,
                       const float* __restrict__ W_hh,
                       const float* __restrict__ b_ih,
                       const float* __restrict__ b_hh,
                       const float* __restrict__ W_fc,
                       const float* __restrict__ b_fc,
                       float* __restrict__ out) {
  const int tid   = threadIdx.x;
  const int bbase = blockIdx.x * ROWS_PER_BLOCK;
  const int b     = bbase + tid;

  // Uniform weights -> SGPRs. Flat gate-row layout: rows 0,1=i 2,3=f 4,5=g 6,7=o.
  // i/f/o rows pre-scaled by 0.5 so sigmoid needs no inner multiply.
  v2f wih[4], bias[4], whh0[4], whh1[4];
#pragma unroll
  for (int g = 0; g < 4; ++g) {
    const float sc = (g == 2) ? 1.0f : 0.5f;     // g-gate (tanh) unscaled
#pragma unroll
    for (int j = 0; j < 2; ++j) {
      const int r = 2 * g + j;
      wih[g][j]  = sc * W_ih[r];
      bias[g][j] = sc * (b_ih[r] + b_hh[r]);
      whh0[g][j] = sc * W_hh[2 * r + 0];
      whh1[g][j] = sc * W_hh[2 * r + 1];
    }
  }

  const float* xr = x + (size_t)b * T_LEN;

  State s;
  s.h = (v2f){0.f, 0.f};
  s.c = (v2f){0.f, 0.f};

  // Double-buffered register pipeline: loads for chunk k+1 in flight while
  // computing chunk k (only ~1 wave/SIMD, so latency must be hidden in-wave).
  float bufA[CHUNK], bufB[CHUNK];
  load_chunk(bufA, xr, 0);
#pragma unroll 1
  for (int t0 = 0; t0 < T_LEN; t0 += 2 * CHUNK) {
    load_chunk(bufB, xr, t0 + CHUNK);
    if (t0 + 5 * CHUNK < T_LEN) {
      __builtin_prefetch(xr + t0 + 4 * CHUNK, 0, 0);   // global_prefetch_b8
      __builtin_prefetch(xr + t0 + 5 * CHUNK, 0, 0);
    }
    lstm_steps(bufA, wih, bias, whh0, whh1, s);
    if (t0 + 2 * CHUNK < T_LEN) load_chunk(bufA, xr, t0 + 2 * CHUNK);
    lstm_steps(bufB, wih, bias, whh0, whh1, s);
  }

  // ---- Epilogue FC: out[b,:] = h @ W_fc^T + b_fc, via WMMA f32 16x16x4 ----
  __shared__ v2f hbuf[ROWS_PER_BLOCK];
  hbuf[tid] = s.h;
  __syncthreads();

  const int lane = tid & 31;
  const int wv   = tid >> 5;        // wave id in block: 0..1
  const int half = lane >> 4;       // 0: K=0,1   1: K=2,3 (zero padding)
  const int n    = lane & 15;       // column / row-in-tile index
  const float kmask = (half == 0) ? 1.0f : 0.0f;
  const float nmask = (n < NCLS) ? 1.0f : 0.0f;

#if __has_builtin(__builtin_amdgcn_wmma_f32_16x16x4_f32)
  // B matrix (4x16): B[k][n] = W_fc[n][k] for n<4, K rows 2,3 zero-padded.
  const int nc = (n < NCLS) ? n : 0;            // clamped -> unconditional load
  v2f bmat;
  bmat.x = kmask * nmask * W_fc[nc * HID + 0];  // K=0 (K=2 -> 0)
  bmat.y = kmask * nmask * W_fc[nc * HID + 1];  // K=1 (K=3 -> 0)
  const float bfc = nmask * b_fc[nc];

#pragma unroll
  for (int tIdx = wv * 2; tIdx < wv * 2 + 2; ++tIdx) {
    // A (16x4): lanes 0-15 -> M=lane, K=0,1 ; lanes 16-31 -> K=2,3 (zeros).
    v2f hval = hbuf[tIdx * 16 + n];             // unconditional ds_load_b64
    v2f a;
    a.x = kmask * hval.x;
    a.y = kmask * hval.y;
    v8f c;
#pragma unroll
    for (int j = 0; j < 8; ++j) c[j] = bfc;     // bias preloaded into accumulator
    v8f d = __builtin_amdgcn_wmma_f32_16x16x4_f32(
        /*neg_a=*/false, a, /*neg_b=*/false, bmat,
        /*c_mod=*/(short)0, c, /*reuse_a=*/false, /*reuse_b=*/false);
    if (n < NCLS) {
      const int rbase = bbase + tIdx * 16 + (half ? 8 : 0);
#pragma unroll
      for (int j = 0; j < 8; ++j)               // VGPR j -> rows M=j / M=j+8
        out[(size_t)(rbase + j) * NCLS + n] = d[j];
    }
  }
#else
  // Scalar fallback (should not be taken on gfx1250).
#pragma unroll
  for (int nn = 0; nn < NCLS; ++nn)
    out[(size_t)b * NCLS + nn] =
        fmaf(s.h.y, W_fc[nn * HID + 1], fmaf(s.h.x, W_fc[nn * HID + 0], b_fc[nn]));
#endif
}

extern "C" void kernel_launch(void* const* d_in, const int* in_sizes, int n_in,
                              void* d_out, int out_size, void* d_ws, size_t ws_size,
                              hipStream_t stream) {
  const float* x    = (const float*)d_in[0];
  const float* W_ih = (const float*)d_in[1];
  const float* W_hh = (const float*)d_in[2];
  const float* b_ih = (const float*)d_in[3];
  const float* b_hh = (const float*)d_in[4];
  const float* W_fc = (const float*)d_in[5];
  const float* b_fc = (const float*)d_in[6];
  float* out = (float*)d_out;

  const int B = in_sizes[0] / T_LEN;          // I == 1
  const int grid = B / ROWS_PER_BLOCK;        // 16384/64 = 256 blocks

  lstm_fused_kernel<<<dim3(grid), dim3(ROWS_PER_BLOCK), 0, stream>>>(
      x, W_ih, W_hh, b_ih, b_hh, W_fc, b_fc, out);
}